// ConditionalDetrTransformer_33990371180996
// MI455X (gfx1250) — compile-verified
//
#include <hip/hip_runtime.h>
#include <hip/hip_bf16.h>
#include <stdint.h>

// Problem constants (from reference)
#define BS   4
#define C    256
#define H    32
#define W    32
#define HW   (H * W)      // 1024
#define NQ   300
#define NCLS 80
#define TOPK (NQ / 2)     // 150

// ---------------------------------------------------------------------------
// Kernel 1: per-batch class-max + exact top-k (stable descending rank) + box
// scaling.  One block per batch, 512 threads (16 wave32).
// ---------------------------------------------------------------------------
__global__ void topk_boxes_kernel(const float* __restrict__ ocls,   // (BS,NQ,NCLS)
                                  const float* __restrict__ coord,  // (BS,NQ,4)
                                  const int*   __restrict__ tsz,    // (BS,2)
                                  float* __restrict__ boxes)        // (BS,TOPK,4)
{
    __shared__ float cls[NQ];
    const int b = blockIdx.x;
    const int t = threadIdx.x;

    if (t < NQ) {
        const float* oc = ocls + (size_t)(b * NQ + t) * NCLS;
        float m = oc[0];
        #pragma unroll 4
        for (int i = 1; i < NCLS; ++i) m = fmaxf(m, oc[i]);
        cls[t] = m;
    }
    __syncthreads();

    if (t < NQ) {
        const float v = cls[t];
        int rank = 0;
        for (int q = 0; q < NQ; ++q) {
            const float u = cls[q];
            rank += (u > v) || (u == v && q < t);   // stable descending rank
        }
        if (rank < TOPK) {
            const float* c4 = coord + (size_t)(b * NQ + t) * 4;
            const float cx = c4[0], cy = c4[1], bw = c4[2], bh = c4[3];
            const float ts0 = (float)tsz[b * 2 + 0];
            const float ts1 = (float)tsz[b * 2 + 1];
            float* o = boxes + (size_t)(b * TOPK + rank) * 4;
            o[0] = ts0 * (cx - 0.5f * bw);   // x1
            o[1] = ts1 * (cy - 0.5f * bh);   // y1
            o[2] = ts0 * (cx + 0.5f * bw);   // x2
            o[3] = ts1 * (cy + 0.5f * bh);   // y2
        }
    }
}

// ---------------------------------------------------------------------------
// Kernel 2: object mask at the 32x32 subsampled grid points.
// gy = i*batched_h/H, gx = j*batched_w/W (integer div, matches arange*bh//h).
// ---------------------------------------------------------------------------
__global__ void mask_kernel(const float* __restrict__ boxes,          // (BS,TOPK,4)
                            const unsigned char* __restrict__ pmask,  // (BS,HW) bool
                            const int* __restrict__ pbh,
                            const int* __restrict__ pbw,
                            int* __restrict__ objmask)                // (BS,HW)
{
    __shared__ float sb[TOPK * 4];
    const int b = blockIdx.x;
    const int p = threadIdx.x;           // 0..1023
    if (p < TOPK * 4) sb[p] = boxes[(size_t)b * TOPK * 4 + p];
    __syncthreads();

    const int i = p >> 5;                // row in H
    const int j = p & 31;                // col in W
    const int bh = pbh[0], bw = pbw[0];
    const float gy = (float)((i * bh) / H);
    const float gx = (float)((j * bw) / W);

    bool any = false;
    #pragma unroll 2
    for (int k = 0; k < TOPK; ++k) {
        const float x1 = sb[4 * k + 0], y1 = sb[4 * k + 1];
        const float x2 = sb[4 * k + 2], y2 = sb[4 * k + 3];
        any = any || ((gx > x1) && (gx < x2) && (gy > y1) && (gy < y2));
    }
    const bool extra = !any;
    const bool pad = pmask[(size_t)b * HW + p] != 0;
    objmask[(size_t)b * HW + p] = (extra || pad) ? 1 : 0;
}

// ---------------------------------------------------------------------------
// Kernel 3: stable partition via inclusive scan (Hillis-Steele in LDS).
// srcidx[b][rank] = p for masked pixels in increasing p; objnum[b] = count.
// ---------------------------------------------------------------------------
__global__ void scan_kernel(const int* __restrict__ objmask,
                            int* __restrict__ srcidx,
                            int* __restrict__ objnum)
{
    __shared__ int sc[HW];
    const int b = blockIdx.x;
    const int p = threadIdx.x;
    const int m = objmask[(size_t)b * HW + p];
    sc[p] = m;
    __syncthreads();
    for (int off = 1; off < HW; off <<= 1) {
        int v = sc[p];
        if (p >= off) v += sc[p - off];
        __syncthreads();
        sc[p] = v;
        __syncthreads();
    }
    const int incl = sc[p];
    const int total = sc[HW - 1];
    if (m) srcidx[(size_t)b * HW + (incl - m)] = p;
    if (p == 0) objnum[b] = total;
}

// ---------------------------------------------------------------------------
// Kernel 4: the bandwidth kernel.  One block per output row-pair (b,pos).
// Channel gather (stride 4KB) is done with per-lane async global->LDS B32
// (CDNA5 data mover, ASYNCcnt), then the contiguous 1KB output rows are
// streamed with async LDS->global B128 stores.  Invalid rows are zero-filled.
// ---------------------------------------------------------------------------
__global__ void gather_kernel(const float* __restrict__ x,    // (BS,C,HW)
                              const float* __restrict__ pe,   // (BS,C,HW)
                              const int* __restrict__ srcidx, // (BS,HW)
                              const int* __restrict__ objnum, // (BS)
                              float* __restrict__ out)        // 2 x (HW,BS,C)
{
    __shared__ __align__(16) float lds[2 * C];
    const int t   = threadIdx.x;          // 0..255  (channel)
    const int bp  = blockIdx.x;           // 0..4095
    const int b   = bp >> 10;
    const int pos = bp & (HW - 1);

    float* outx = out + (size_t)(pos * BS + b) * C;
    float* outp = outx + (size_t)HW * BS * C;

    const int on = objnum[b];
    if (pos >= on) {
        outx[t] = 0.0f;
        outp[t] = 0.0f;
        return;
    }

    const int src = srcidx[(size_t)b * HW + pos];

    // Per-lane gather: lane t copies x at (b, t, src) to LDS slot t,   (addresses stride 4KB)
    //                  and pe at (b, t, src) to LDS slot C+t
    const unsigned long long gx =
        (unsigned long long)(size_t)(x  + ((size_t)(b * C + t)) * HW + src);
    const unsigned long long gp =
        (unsigned long long)(size_t)(pe + ((size_t)(b * C + t)) * HW + src);
    const unsigned lx = (unsigned)(size_t)&lds[t];
    const unsigned lp = (unsigned)(size_t)&lds[C + t];

    asm volatile("global_load_async_to_lds_b32 %0, %1, off"
                 :: "v"(lx), "v"(gx) : "memory");
    asm volatile("global_load_async_to_lds_b32 %0, %1, off"
                 :: "v"(lp), "v"(gp) : "memory");
    asm volatile("s_wait_asynccnt 0" ::: "memory");
    __syncthreads();

    // Stream the two contiguous 1KB rows out: 64 lanes x 16B each.
    if (t < 128) {
        const bool isx = (t < 64);
        const int lane = isx ? t : (t - 64);
        const unsigned lsrc = (unsigned)(size_t)&lds[(isx ? 0 : C) + lane * 4];
        float* gdst = (isx ? outx : outp) + lane * 4;
        asm volatile("global_store_async_from_lds_b128 %0, %1, off"
                     :: "v"((unsigned long long)(size_t)gdst), "v"(lsrc)
                     : "memory");
    }
    asm volatile("s_wait_asynccnt 0" ::: "memory");
}

// ---------------------------------------------------------------------------
// Launcher
// ---------------------------------------------------------------------------
extern "C" void kernel_launch(void* const* d_in, const int* in_sizes, int n_in,
                              void* d_out, int out_size, void* d_ws, size_t ws_size,
                              hipStream_t stream) {
    const float*         x     = (const float*)d_in[0];
    const float*         pe    = (const float*)d_in[1];
    const unsigned char* mask  = (const unsigned char*)d_in[2];
    const float*         coord = (const float*)d_in[3];
    const float*         ocls  = (const float*)d_in[4];
    const int*           tsz   = (const int*)d_in[5];
    const int*           pbh   = (const int*)d_in[6];
    const int*           pbw   = (const int*)d_in[7];
    float*               out   = (float*)d_out;

    // Workspace layout (~42 KB)
    char* ws = (char*)d_ws;
    float* boxes  = (float*)ws;                               // BS*TOPK*4 floats
    int*   objm   = (int*)(ws + sizeof(float) * BS * TOPK * 4);
    int*   srcidx = objm + BS * HW;
    int*   objnum = srcidx + BS * HW;

    topk_boxes_kernel<<<dim3(BS), dim3(512), 0, stream>>>(ocls, coord, tsz, boxes);
    mask_kernel<<<dim3(BS), dim3(HW), 0, stream>>>(boxes, mask, pbh, pbw, objm);
    scan_kernel<<<dim3(BS), dim3(HW), 0, stream>>>(objm, srcidx, objnum);
    gather_kernel<<<dim3(BS * HW), dim3(C), 0, stream>>>(x, pe, srcidx, objnum, out);
}